// SSIMLoss_22436909154727
// MI455X (gfx1250) — compile-verified
//
#include <hip/hip_runtime.h>

// ---------------------------------------------------------------------------
// SSIM loss for 32x3x512x512 f32 images on gfx1250 (MI455X).
// Separable 11-tap Gaussian depthwise conv as banded matmuls on
// V_WMMA_F32_16X16X4_F32 (wave32 WMMA). Image tiles staged into LDS by the
// Tensor Data Mover (tensor_load_to_lds + s_wait_tensorcnt) when available,
// falling back to VGPR staging otherwise. Per-wave LDS relayout scratch
// between horizontal and vertical conv passes.
// ---------------------------------------------------------------------------

typedef float v2f __attribute__((ext_vector_type(2)));
typedef float v8f __attribute__((ext_vector_type(8)));

#define NPLANE 96          // 32 batch * 3 channels
#define IMH    512
#define IMW    512
#define OH     502         // valid conv output
#define OW     502
#define TPITCH 76          // LDS image tile pitch (= TDM tile width, floats)
#define TROWS  48          // LDS image tile rows (32 out rows + 16 WMMA slack)

// Normalized 11-tap Gaussian, sigma = 1.5 (matches reference _make_window)
__constant__ float c_gw[11] = {
    0.00102838f, 0.00759875f, 0.03600077f, 0.10936069f, 0.21300553f,
    0.26601172f,
    0.21300553f, 0.10936069f, 0.03600077f, 0.00759875f, 0.00102838f
};

// Band matrix element: B[j][i] = w[j-i] when 0 <= j-i <= 10, else 0.
__device__ __forceinline__ float bandw(int j, int i) {
    int d = j - i;
    return (d >= 0 && d <= 10) ? c_gw[d] : 0.0f;
}

__device__ __forceinline__ v8f zero8() {
    v8f z = {0.f, 0.f, 0.f, 0.f, 0.f, 0.f, 0.f, 0.f};
    return z;
}

// ---------------------------------------------------------------------------
// Tensor Data Mover staging (CDNA5).  2D tile load, descriptor per ISA
// CDNA5 ISA.  OOB reads return zero, which is safe
// here: all zero-filled slack feeds zero band coefficients or masked outputs.
// ---------------------------------------------------------------------------
#if __has_builtin(__builtin_amdgcn_tensor_load_to_lds)
#define SSIM_HAVE_TDM 1
#warning SSIM_PROBE: TDM path ACTIVE (__builtin_amdgcn_tensor_load_to_lds found)

typedef unsigned int u32x4 __attribute__((ext_vector_type(4)));
typedef int          i32x4 __attribute__((ext_vector_type(4)));
typedef int          i32x8 __attribute__((ext_vector_type(8)));

__device__ __forceinline__ void tdm_load_tile_2d(
    const float* gsrc,      // global address of tile origin
    unsigned lds_off,       // LDS byte offset of destination
    int rem_w, int rem_h,   // remaining tensor extent from origin (elements)
    int tile_w, int tile_h, // tile dims (elements)
    int row_stride)         // tensor dim0 stride (elements)
{
    unsigned long long ga = (unsigned long long)(uintptr_t)gsrc;

    u32x4 g0;
    g0.x = 1u;                                            // count=1, user mode
    g0.y = lds_off;                                       // lds_addr (bytes)
    g0.z = (unsigned)(ga & 0xffffffffu);                  // global_addr[31:0]
    g0.w = (unsigned)((ga >> 32) & 0x01ffffffu)           // global_addr[56:32]
         | (2u << 30);                                    // type = 2 ("image")

    unsigned rw = (unsigned)rem_w, rh = (unsigned)rem_h;
    i32x8 g1;
    g1[0] = (int)(2u << 16);                              // data_size = 4B
    g1[1] = (int)(rw << 16);                              // tensor_dim0[15:0]
    g1[2] = (int)((rw >> 16) | (rh << 16));               // dim0[31:16]|dim1[15:0]
    g1[3] = (int)((rh >> 16) | ((unsigned)tile_w << 16)); // dim1[31:16]|tile_dim0
    g1[4] = (int)((unsigned)tile_h & 0xffffu);            // tile_dim1; tile_dim2=0
    g1[5] = row_stride;                                   // dim0_stride[31:0]
    g1[6] = 0;                                            // stride[47:32]|d1s[15:0]
    g1[7] = 0;                                            // dim1_stride[47:16]

    i32x4 z4 = {0, 0, 0, 0};
#if __has_include(<hip/amd_detail/amd_gfx1250_TDM.h>)
#warning SSIM_PROBE: TDM 6-arg variant (amdgpu-toolchain headers present)
    // amdgpu-toolchain (clang-23 / therock-10.0): 6-arg builtin
    i32x8 z8 = {0, 0, 0, 0, 0, 0, 0, 0};
    __builtin_amdgcn_tensor_load_to_lds(g0, g1, z4, z4, z8, 0);
#else
#warning SSIM_PROBE: TDM 5-arg variant (ROCm clang-22 form)
    // ROCm 7.2 (clang-22): 5-arg builtin
    __builtin_amdgcn_tensor_load_to_lds(g0, g1, z4, z4, 0);
#endif
}
#else
#warning SSIM_PROBE: TDM builtin NOT found -- VGPR staging fallback in use
#endif // __has_builtin(tensor_load_to_lds)

// ---------------------------------------------------------------------------
// ws layout: ws[0] = float sum accumulator; ((int*)ws)[1] = max>128 flag;
//            ((int*)ws)[2] = min<-0.5 flag.
// ---------------------------------------------------------------------------

__global__ void ssim_init(float* ws) {
    ws[0] = 0.0f;
    ((int*)ws)[1] = 0;
    ((int*)ws)[2] = 0;
}

// Dynamic-range heuristic flags over img1 (grid-stride, float4 vectorized).
__global__ void ssim_range_flags(const float4* __restrict__ img1, int n4,
                                 float* ws) {
    int stride = gridDim.x * blockDim.x;
    bool gt = false, lt = false;
    for (int i = blockIdx.x * blockDim.x + threadIdx.x; i < n4; i += stride) {
        float4 v = img1[i];
        gt = gt || (v.x > 128.0f) || (v.y > 128.0f) ||
                   (v.z > 128.0f) || (v.w > 128.0f);
        lt = lt || (v.x < -0.5f) || (v.y < -0.5f) ||
                   (v.z < -0.5f) || (v.w < -0.5f);
    }
    if (gt) atomicOr(((int*)ws) + 1, 1);
    if (lt) atomicOr(((int*)ws) + 2, 1);
}

// ---------------------------------------------------------------------------
// Main kernel: one block = one plane sub-region of 32(out rows) x 64(out cols),
// 8 waves, each wave owns a 16x16 output tile.
// ---------------------------------------------------------------------------
__global__ __launch_bounds__(256)
void ssim_map_kernel(const float* __restrict__ img1,
                     const float* __restrict__ img2,
                     float* ws) {
    __shared__ float lds1[TROWS * TPITCH];     // img1 tile
    __shared__ float lds2[TROWS * TPITCH];     // img2 tile
    __shared__ float tbuf[8 * 32 * 16];        // per-wave horizontal scratch

    const int tid  = threadIdx.x;
    const int lane = tid & 31;
    const int wv   = tid >> 5;                 // wave id 0..7
    const int half = lane >> 4;                // 0: lanes 0-15, 1: lanes 16-31
    const int m    = lane & 15;

    const int bid   = blockIdx.x;
    const int plane = bid >> 7;                // 128 blocks per plane
    const int rem   = bid & 127;
    const int oy0   = (rem >> 3) * 32;         // 16 row-blocks of 32
    const int ox0   = (rem & 7) * 64;          // 8 col-blocks of 64

    const float* p1 = img1 + (size_t)plane * (IMH * IMW);
    const float* p2 = img2 + (size_t)plane * (IMH * IMW);

    // ---- stage image tiles into LDS ----
#if SSIM_HAVE_TDM
    if (tid == 0) {
        const float* t1 = p1 + (size_t)oy0 * IMW + ox0;
        const float* t2 = p2 + (size_t)oy0 * IMW + ox0;
        int rw = IMW - ox0, rh = IMH - oy0;    // OOB beyond this reads zero
        tdm_load_tile_2d(t1, (unsigned)(uintptr_t)(void*)lds1,
                         rw, rh, TPITCH, TROWS, IMW);
        tdm_load_tile_2d(t2, (unsigned)(uintptr_t)(void*)lds2,
                         rw, rh, TPITCH, TROWS, IMW);
        __builtin_amdgcn_s_wait_tensorcnt(0);
    }
    __syncthreads();
#else
    for (int idx = tid; idx < TROWS * TPITCH; idx += 256) {
        int r  = idx / TPITCH;
        int c  = idx - r * TPITCH;
        int gy = oy0 + r;
        int gx = ox0 + c;
        bool in = (gy < IMH) && (gx < IMW);
        size_t g = (size_t)min(gy, IMH - 1) * IMW + min(gx, IMW - 1);
        lds1[idx] = in ? p1[g] : 0.0f;         // zero-fill to match TDM OOB
        lds2[idx] = in ? p2[g] : 0.0f;
    }
    __syncthreads();
#endif

    // ---- per-lane coefficient fragments (identical for A- and B-side band) --
    // A 16x4 f32 layout: lanes 0-15 hold K=kb+{0,1}; lanes 16-31 hold K=kb+{2,3}
    // B  4x16 f32 layout: vreg0 = rows kb / kb+2 across lane halves, vreg1 = +1
    v2f wc[7];
#pragma unroll
    for (int c = 0; c < 7; ++c) {
        int k = 4 * c + 2 * half;
        wc[c].x = bandw(k, m);
        wc[c].y = bandw(k + 1, m);
    }

    const int row0 = (wv >> 2) * 16;           // subtile row offset (0 or 16)
    const int col0 = (wv & 3) * 16;            // subtile col offset (0..48)
    float* tw = &tbuf[wv * 512];               // 32 rows x 16 cols scratch

    v8f res[5];

#pragma unroll
    for (int q = 0; q < 5; ++q) {              // x, y, x^2, y^2, xy
        // ---- horizontal pass: C[r][x] = sum_k In[r][col0+k] * band[k][x] ----
        v8f h0 = zero8();
        v8f h1 = zero8();
#pragma unroll
        for (int c = 0; c < 7; ++c) {
            int k0 = 4 * c + 2 * half;
#pragma unroll
            for (int rg = 0; rg < 2; ++rg) {
                int rr   = row0 + rg * 16 + m;
                int base = rr * TPITCH + col0 + k0;
                float x0 = lds1[base], x1 = lds1[base + 1];
                float y0 = lds2[base], y1 = lds2[base + 1];
                v2f a;
                if (q == 0)      { a.x = x0;        a.y = x1; }
                else if (q == 1) { a.x = y0;        a.y = y1; }
                else if (q == 2) { a.x = x0 * x0;   a.y = x1 * x1; }
                else if (q == 3) { a.x = y0 * y0;   a.y = y1 * y1; }
                else             { a.x = x0 * y0;   a.y = x1 * y1; }
                if (rg == 0)
                    h0 = __builtin_amdgcn_wmma_f32_16x16x4_f32(
                        false, a, false, wc[c], (short)0, h0, false, false);
                else
                    h1 = __builtin_amdgcn_wmma_f32_16x16x4_f32(
                        false, a, false, wc[c], (short)0, h1, false, false);
            }
        }

        // ---- relayout C(16x16 rows-in-vgprs) -> LDS (32 rows x 16 cols) ----
#pragma unroll
        for (int v = 0; v < 8; ++v) {
            tw[(v + half * 8) * 16 + m]      = h0[v];
            tw[(v + half * 8 + 16) * 16 + m] = h1[v];
        }
        // same-wave DS ordering guarantees store->load visibility

        // ---- vertical pass: C[y][x] = sum_j band[j][y]^T * T[j][x] ----
        v8f acc = zero8();
#pragma unroll
        for (int c = 0; c < 7; ++c) {
            int k0 = 4 * c + 2 * half;
            v2f b;
            b.x = tw[k0 * 16 + m];
            b.y = tw[(k0 + 1) * 16 + m];
            acc = __builtin_amdgcn_wmma_f32_16x16x4_f32(
                false, wc[c], false, b, (short)0, acc, false, false);
        }
        res[q] = acc;
    }

    // ---- SSIM map + masked accumulation ----
    int fmax = ((const int*)ws)[1];
    int fmin = ((const int*)ws)[2];
    float maxv = fmax ? 255.0f : 1.0f;
    float minv = fmin ? -1.0f : 0.0f;
    float L  = maxv - minv;
    float C1 = (0.01f * L) * (0.01f * L);
    float C2 = (0.03f * L) * (0.03f * L);

    float sum = 0.0f;
#pragma unroll
    for (int v = 0; v < 8; ++v) {
        int gy = oy0 + row0 + half * 8 + v;
        int gx = ox0 + col0 + m;
        float mu1 = res[0][v], mu2 = res[1][v];
        float ex2 = res[2][v], ey2 = res[3][v], exy = res[4][v];
        float mu1s = mu1 * mu1, mu2s = mu2 * mu2, m12 = mu1 * mu2;
        float s1 = ex2 - mu1s, s2 = ey2 - mu2s, s12 = exy - m12;
        float num = (2.0f * m12 + C1) * (2.0f * s12 + C2);
        float den = (mu1s + mu2s + C1) * (s1 + s2 + C2);
        if (gy < OH && gx < OW) sum += num / den;
    }

    // wave32 reduction, one atomic per wave
#pragma unroll
    for (int off = 16; off > 0; off >>= 1)
        sum += __shfl_down(sum, off, 32);
    if (lane == 0) atomicAdd(ws, sum);
}

__global__ void ssim_finalize(const float* ws, float* out) {
    out[0] = 1.0f - ws[0] * (1.0f / (float)(NPLANE * OH * OW));
}

// ---------------------------------------------------------------------------

extern "C" void kernel_launch(void* const* d_in, const int* in_sizes, int n_in,
                              void* d_out, int out_size, void* d_ws,
                              size_t ws_size, hipStream_t stream) {
    const float* img1 = (const float*)d_in[0];
    const float* img2 = (const float*)d_in[1];
    float* out = (float*)d_out;
    float* ws  = (float*)d_ws;

    hipLaunchKernelGGL(ssim_init, dim3(1), dim3(1), 0, stream, ws);

    int n4 = in_sizes[0] / 4;
    hipLaunchKernelGGL(ssim_range_flags, dim3(2048), dim3(256), 0, stream,
                       (const float4*)img1, n4, ws);

    // 96 planes * (16 row-blocks of 32) * (8 col-blocks of 64) = 12288 blocks
    hipLaunchKernelGGL(ssim_map_kernel, dim3(NPLANE * 16 * 8), dim3(256), 0,
                       stream, img1, img2, ws);

    hipLaunchKernelGGL(ssim_finalize, dim3(1), dim3(1), 0, stream, ws, out);
}